// ConvectionErosionPdeM2_18880676233736
// MI455X (gfx1250) — compile-verified
//
#include <hip/hip_runtime.h>
#include <math.h>

// Problem constants (match the reference)
#define B_    4
#define CH_   8
#define OR_   8
#define H_    160
#define W_    160
#define KOR_  3
#define KH_   5
#define KW_   5
#define NOFF  (KOR_*KH_*KW_)   // 75
#define PAD   4
#define HP    (H_ + 2*PAD)     // 168
#define WP    (W_ + 2*PAD)     // 168
#define TILE  32
#define TH_T  (TILE + 2*PAD)   // 40
#define TW_T  (TILE + 2*PAD)   // 40
#define PI_F  3.14159265358979323846

typedef __attribute__((ext_vector_type(4))) unsigned int u32x4;
typedef __attribute__((ext_vector_type(8))) int          i32x8;
typedef __attribute__((ext_vector_type(4))) int          i32x4;

// ---------------------------------------------------------------------------
// Kernel 1: convect -> padded workspace v[B][CH][OR][HP][WP]
// v_pad[py][px] = trilerp(input[b,c], tc, yc, xc) at the clamped pixel
// (y,x) = (clip(py-4), clip(px-4)).  Pad cells keep the TDM tile fetch
// in-bounds; erode never reads them through its clamped-index path.
// ---------------------------------------------------------------------------
__global__ __launch_bounds__(256)
void convect_kernel(const float* __restrict__ in,
                    const float* __restrict__ g0,
                    float* __restrict__ v)
{
    int id = blockIdx.x * blockDim.x + threadIdx.x;
    const int total = B_ * CH_ * OR_ * HP * WP;
    if (id >= total) return;

    int px = id % WP;  int r  = id / WP;
    int py = r  % HP;  r /= HP;
    int to = r  % OR_; r /= OR_;
    int c  = r  % CH_;
    int b  = r  / CH_;

    int x = min(max(px - PAD, 0), W_ - 1);
    int y = min(max(py - PAD, 0), H_ - 1);

    float gx0 = g0[c*3 + 0], gy0 = g0[c*3 + 1], th0 = g0[c*3 + 2];
    float c0 = cosf(th0), s0 = sinf(th0);
    float tix = -(c0*gx0 + s0*gy0);
    float tiy =  s0*gx0 - c0*gy0;

    float th  = (float)to * (float)(PI_F/4.0);
    float cth = cosf(th), sth = sinf(th);

    float xc = (float)x + cth*tix - sth*tiy;
    float yc = (float)y + sth*tix + cth*tiy;
    float tc = (float)to - th0 * (float)(OR_ / (2.0*PI_F));

    float tf = floorf(tc), yf = floorf(yc), xf = floorf(xc);
    float wt = tc - tf, wy = yc - yf, wx = xc - xf;
    int t0 = ((int)tf % OR_ + OR_) % OR_;
    int t1 = (t0 + 1) & (OR_ - 1);
    int y0 = min(max((int)yf, 0), H_ - 1);
    int y1 = min(y0 + 1, H_ - 1);
    int x0 = min(max((int)xf, 0), W_ - 1);
    int x1 = min(x0 + 1, W_ - 1);

    const float* base = in + ((size_t)(b*CH_ + c)) * (OR_*H_*W_);
#define G_(tt,yy,xx) base[((size_t)(tt)*H_ + (yy)) * W_ + (xx)]
    float s =
        (1.f-wt) * ( (1.f-wy) * ((1.f-wx)*G_(t0,y0,x0) + wx*G_(t0,y0,x1))
                   +      wy  * ((1.f-wx)*G_(t0,y1,x0) + wx*G_(t0,y1,x1)) )
      +      wt  * ( (1.f-wy) * ((1.f-wx)*G_(t1,y0,x0) + wx*G_(t1,y0,x1))
                   +      wy  * ((1.f-wx)*G_(t1,y1,x0) + wx*G_(t1,y1,x1)) );
#undef G_

    v[ (((size_t)(b*CH_ + c))*OR_ + to) * (HP*WP) + (size_t)py*WP + px ] = s;
}

// ---------------------------------------------------------------------------
// Erode accumulation: one thread produces 4 x-consecutive outputs for one
// orientation `to`.  Weights/offsets are uniform per (t,k); interior tiles use
// 2x5 contiguous taps shared across the 4 outputs (2.5 LDS loads/output);
// border tiles apply the reference clip rule per output.
// ---------------------------------------------------------------------------
template <bool BORDER>
__device__ __forceinline__
void erode_accum4(const float* __restrict__ tile,   // [OR_][TH_T][TW_T]
                  const float* __restrict__ kv_s,   // [NOFF]
                  int to, int ly, int lx, int Y, int X, int ty0, int tx0,
                  float acc[4])
{
    float th  = (float)to * (float)(PI_F/4.0);
    float cth = cosf(th), sth = sinf(th);
    acc[0] = acc[1] = acc[2] = acc[3] = INFINITY;
    int k = 0;
    #pragma unroll
    for (int dt = -1; dt <= 1; ++dt) {
        int tk = (to + dt + OR_) & (OR_ - 1);
        const float* plane = tile + (size_t)tk * (TH_T * TW_T);
        #pragma unroll
        for (int dy = -2; dy <= 2; ++dy) {
            #pragma unroll
            for (int dx = -2; dx <= 2; ++dx, ++k) {
                float dxf = (float)dx, dyf = (float)dy;
                float ox = cth*dxf - sth*dyf;
                float oy = sth*dxf + cth*dyf;
                float oxf = floorf(ox), oyf = floorf(oy);
                float wx = ox - oxf, wy = oy - oyf;
                float omwx = 1.f - wx, omwy = 1.f - wy;
                int ix = (int)oxf, iy = (int)oyf;
                float kvk = kv_s[k];
                if (!BORDER) {
                    int r0 = ly + PAD + iy;
                    int c0 = lx + PAD + ix;
                    const float* p0 = plane + (r0    )*TW_T + c0;
                    const float* p1 = plane + (r0 + 1)*TW_T + c0;
                    float a0=p0[0], a1=p0[1], a2=p0[2], a3=p0[3], a4=p0[4];
                    float b0=p1[0], b1=p1[1], b2=p1[2], b3=p1[3], b4=p1[4];
                    float h00 = omwx*a0 + wx*a1;
                    float h01 = omwx*a1 + wx*a2;
                    float h02 = omwx*a2 + wx*a3;
                    float h03 = omwx*a3 + wx*a4;
                    float h10 = omwx*b0 + wx*b1;
                    float h11 = omwx*b1 + wx*b2;
                    float h12 = omwx*b2 + wx*b3;
                    float h13 = omwx*b3 + wx*b4;
                    acc[0] = fminf(acc[0], (omwy*h00 + wy*h10) + kvk);
                    acc[1] = fminf(acc[1], (omwy*h01 + wy*h11) + kvk);
                    acc[2] = fminf(acc[2], (omwy*h02 + wy*h12) + kvk);
                    acc[3] = fminf(acc[3], (omwy*h03 + wy*h13) + kvk);
                } else {
                    int gy0 = min(max(Y + iy, 0), H_ - 1);
                    int gy1 = min(gy0 + 1, H_ - 1);
                    int r0 = gy0 - ty0 + PAD, r1 = gy1 - ty0 + PAD;
                    #pragma unroll
                    for (int j = 0; j < 4; ++j) {
                        int gx0 = min(max(X + j + ix, 0), W_ - 1);
                        int gx1 = min(gx0 + 1, W_ - 1);
                        int c0 = gx0 - tx0 + PAD, c1 = gx1 - tx0 + PAD;
                        float v00 = plane[r0*TW_T + c0];
                        float v01 = plane[r0*TW_T + c1];
                        float v10 = plane[r1*TW_T + c0];
                        float v11 = plane[r1*TW_T + c1];
                        float s = omwy*(omwx*v00 + wx*v01)
                                +  wy *(omwx*v10 + wx*v11);
                        acc[j] = fminf(acc[j], s + kvk);
                    }
                }
            }
        }
    }
}

// ---------------------------------------------------------------------------
// Kernel 2: erode.  One block = 32x32 spatial tile for one (b,c); all 8
// orientations share one TDM-staged (8,40,40) f32 LDS tile (51.2 KB).
// 256 threads = 32 rows x 8 groups of 4 columns.
// ---------------------------------------------------------------------------
__global__ __launch_bounds__(256)
void erode_kernel(const float* __restrict__ vpad,
                  const float* __restrict__ metric,
                  float* __restrict__ out)
{
    __shared__ float tile[OR_ * TH_T * TW_T];   // 51200 B
    __shared__ float kv_s[NOFF];

    const int tid = threadIdx.x;
    const int bx = blockIdx.x, by = blockIdx.y;
    const int bc = blockIdx.z;
    const int c  = bc % CH_;
    const int ty0 = by * TILE, tx0 = bx * TILE;

    // ---- TDM: async 3-D tile load (8 planes x 40 rows x 40 cols of f32) ----
    if (tid < 32) {
        unsigned long long gaddr = (unsigned long long)(uintptr_t)
            (vpad + ((size_t)bc * OR_) * (HP*WP) + (size_t)ty0 * WP + tx0);
        unsigned int lds_addr = (unsigned int)(uintptr_t)(&tile[0]);

        u32x4 g0d;
        g0d[0] = 1u;                                     // count=1, user D#
        g0d[1] = lds_addr;                               // lds_addr
        g0d[2] = (unsigned int)(gaddr & 0xFFFFFFFFull);  // global_addr[31:0]
        g0d[3] = (unsigned int)((gaddr >> 32) & 0x1FFFFFFull) | (2u << 30); // addr[56:32], type=2

        const unsigned s1 = (unsigned)(HP*WP);           // 28224, dim1 stride
        i32x8 g1d;
        g1d[0] = (int)(2u << 16);                        // data_size=4B; mask=0
        g1d[1] = (int)((unsigned)WP << 16);              // tensor_dim0[15:0]=168
        g1d[2] = (int)((unsigned)HP << 16);              // tensor_dim1[15:0]=168
        g1d[3] = (int)((unsigned)TW_T << 16);            // tile_dim0=40
        g1d[4] = (int)((unsigned)TH_T | ((unsigned)OR_ << 16)); // tile_dim1=40, tile_dim2=8
        g1d[5] = (int)(unsigned)WP;                      // tensor_dim0_stride=168
        g1d[6] = (int)((s1 & 0xFFFFu) << 16);            // dim0_stride[47:32]=0 | dim1_stride[15:0]
        g1d[7] = (int)(s1 >> 16);                        // dim1_stride[47:16]=0

        i32x4 g2d; g2d[0] = OR_; g2d[1] = 0; g2d[2] = 0; g2d[3] = 0; // tensor_dim2=8
        i32x4 g3d; g3d[0] = 0;  g3d[1] = 0;  g3d[2] = 0;  g3d[3] = 0;
        i32x8 g4d; g4d[0] = 0;  g4d[1] = 0;  g4d[2] = 0;  g4d[3] = 0;
                   g4d[4] = 0;  g4d[5] = 0;  g4d[6] = 0;  g4d[7] = 0;

        __builtin_amdgcn_tensor_load_to_lds(g0d, g1d, g2d, g3d, g4d, 0);
        __builtin_amdgcn_s_wait_tensorcnt(0);
    }

    // ---- kv table (per-channel), 75 threads ----
    if (tid < NOFF) {
        int k   = tid;
        int dt  = k / (KH_*KW_) - KOR_/2;
        int rem = k % (KH_*KW_);
        int dy  = rem / KW_ - KH_/2;
        int dx  = rem % KW_ - KW_/2;
        double DTH = 2.0*PI_F/OR_;
        double v3 = (double)dt * DTH;
        double h  = v3 * 0.5;
        double cc = (fabs(h) < 1e-6) ? (1.0 - h*h/3.0) : (h / tan(h));
        double v1 =  cc*(double)dx + h*(double)dy;
        double v2 = -h*(double)dx + cc*(double)dy;
        float l1 = (float)v1, l2 = (float)v2, l3 = (float)v3;
        float m0 = metric[c*3+0], m1 = metric[c*3+1], m2 = metric[c*3+2];
        float d2 = (l1*m0)*(l1*m0) + (l2*m1)*(l2*m1) + (l3*m2)*(l3*m2);
        const double A   = 0.65;
        const double Pd  = A / (2.0*A - 1.0);
        const double NUd = (2.0*A - 1.0) * pow(2.0*A, -2.0*A/(2.0*A - 1.0));
        kv_s[k] = (float)NUd * powf(d2, (float)Pd);
    }
    __syncthreads();

    const int lx = (tid & 7) * 4;       // 8 groups of 4 consecutive columns
    const int ly = tid >> 3;            // 32 rows
    const int X = tx0 + lx, Y = ty0 + ly;
    float* outb = out + (size_t)bc * (OR_*H_*W_);

    const bool border = (bx == 0) | (by == 0) |
                        (bx == (W_/TILE - 1)) | (by == (H_/TILE - 1));

    float acc[4];
    if (border) {
        for (int to = 0; to < OR_; ++to) {
            erode_accum4<true >(tile, kv_s, to, ly, lx, Y, X, ty0, tx0, acc);
            float4 o = make_float4(acc[0], acc[1], acc[2], acc[3]);
            *(float4*)(outb + (size_t)to*(H_*W_) + (size_t)Y*W_ + X) = o;
        }
    } else {
        for (int to = 0; to < OR_; ++to) {
            erode_accum4<false>(tile, kv_s, to, ly, lx, Y, X, ty0, tx0, acc);
            float4 o = make_float4(acc[0], acc[1], acc[2], acc[3]);
            *(float4*)(outb + (size_t)to*(H_*W_) + (size_t)Y*W_ + X) = o;
        }
    }
}

// ---------------------------------------------------------------------------
extern "C" void kernel_launch(void* const* d_in, const int* in_sizes, int n_in,
                              void* d_out, int out_size, void* d_ws, size_t ws_size,
                              hipStream_t stream)
{
    const float* inp    = (const float*)d_in[0];   // (B,CH,OR,H,W) f32
    const float* g0     = (const float*)d_in[1];   // (CH,3) f32
    const float* metric = (const float*)d_in[2];   // (CH,3) f32
    float* out  = (float*)d_out;
    float* vpad = (float*)d_ws;                    // B*CH*OR*HP*WP floats (~28.9 MB)

    const int total = B_ * CH_ * OR_ * HP * WP;
    convect_kernel<<<(total + 255)/256, 256, 0, stream>>>(inp, g0, vpad);

    dim3 grid(W_/TILE, H_/TILE, B_*CH_);
    erode_kernel<<<grid, 256, 0, stream>>>(vpad, metric, out);
}